// TopKTopPSampler_81870666596473
// MI455X (gfx1250) — compile-verified
//
#include <hip/hip_runtime.h>
#include <stdint.h>

#define TPB   512
#define CHUNK 2048      // floats per staged chunk = TPB*4
#define NB    2048      // bins for 11-bit radix levels
#define NB3   1024      // bins for final 10-bit level
#define VV    128000

static_assert(CHUNK == 4 * TPB, "chunk sizing");

__device__ __forceinline__ uint32_t f2key(float f) {
  uint32_t u = __float_as_uint(f);
  return (u & 0x80000000u) ? ~u : (u | 0x80000000u);   // ascending-order key
}
__device__ __forceinline__ float key2f(uint32_t k) {
  uint32_t u = (k & 0x80000000u) ? (k ^ 0x80000000u) : ~k;
  return __uint_as_float(u);
}

// Stream one row through LDS using the CDNA5 async global->LDS DMA path,
// double buffered. Each wave consumes exactly the LDS region it loaded, so
// only the wave-local s_wait_asynccnt is needed (no block barriers).
template <class F>
__device__ __forceinline__ void stream_row(const float* __restrict__ row,
                                           float invT, float* sbuf, F f) {
  const int t = threadIdx.x;
  const int nch = (VV + CHUNK - 1) / CHUNK;   // 63
  {
    uint32_t ldsoff = (uint32_t)(uintptr_t)(sbuf + 4 * t);
    const float* g = row + 4 * t;
    asm volatile("global_load_async_to_lds_b128 %0, %1, off"
                 :: "v"(ldsoff), "v"(g) : "memory");
  }
  for (int c = 0; c < nch; ++c) {
    float* cur = sbuf + (c & 1) * CHUNK;
    if (c + 1 < nch) {
      int e = (c + 1) * CHUNK + 4 * t;
      if (e > VV - 4) e = VV - 4;              // clamp (consumer bounds-checks)
      uint32_t ldsoff =
          (uint32_t)(uintptr_t)(sbuf + ((c + 1) & 1) * CHUNK + 4 * t);
      const float* g = row + e;
      asm volatile("global_load_async_to_lds_b128 %0, %1, off"
                   :: "v"(ldsoff), "v"(g) : "memory");
      asm volatile("s_wait_asynccnt 0x1" ::: "memory");
    } else {
      asm volatile("s_wait_asynccnt 0x0" ::: "memory");
    }
    const int base = c * CHUNK + 4 * t;
#pragma unroll
    for (int j = 0; j < 4; ++j) {
      int idx = base + j;
      if (idx < VV) f(cur[4 * t + j] * invT, idx);
    }
  }
}

// Find bin (scanning keys from HIGH to LOW) where cumulative count crosses
// `target`. Outputs: su[1] = selected bin, su[2] = residual target.
__device__ __forceinline__ void cross_count(const uint32_t* hist, int nbins,
                                            uint32_t target, uint32_t* gsm,
                                            uint32_t* su) {
  const int t = threadIdx.x;
  const int ng = nbins >> 5;
  if (t < ng) {
    uint32_t s = 0;
#pragma unroll 4
    for (int j = 0; j < 32; ++j) s += hist[nbins - 1 - (32 * t + j)];
    gsm[t] = s;
  }
  __syncthreads();
  if (t == 0) {
    uint32_t A = 0;
    int g = 0;
    for (; g < ng - 1; ++g) {
      if (A + gsm[g] >= target) break;
      A += gsm[g];
    }
    int sel = nbins - 1 - 32 * g;
    uint32_t Asel = A;
    for (int j = 0; j < 32; ++j) {
      int b = nbins - 1 - (32 * g + j);
      uint32_t c = hist[b];
      if (A + c >= target || j == 31) { sel = b; Asel = A; break; }
      A += c;
    }
    su[1] = (uint32_t)sel;
    su[2] = target - Asel;
  }
  __syncthreads();
}

// Weighted version (float bins). If scaleP, target = p * (total weight S).
// Outputs: su[1] = selected bin, sf[1] = residual weight target.
__device__ __forceinline__ void cross_weight(const float* fh, int nbins,
                                             bool scaleP, float p, float target,
                                             float* gsf, uint32_t* su,
                                             float* sf) {
  const int t = threadIdx.x;
  const int ng = nbins >> 5;
  if (t < ng) {
    float s = 0.f;
#pragma unroll 4
    for (int j = 0; j < 32; ++j) s += fh[nbins - 1 - (32 * t + j)];
    gsf[t] = s;
  }
  __syncthreads();
  if (t == 0) {
    float T = target;
    if (scaleP) {
      float S = 0.f;
      for (int g = 0; g < ng; ++g) S += gsf[g];
      T = p * S;
    }
    float A = 0.f;
    int g = 0;
    for (; g < ng - 1; ++g) {
      if (A + gsf[g] >= T) break;
      A += gsf[g];
    }
    int sel = nbins - 1 - 32 * g;
    float Asel = A;
    for (int j = 0; j < 32; ++j) {
      int b = nbins - 1 - (32 * g + j);
      float c = fh[b];
      if (A + c >= T || j == 31) { sel = b; Asel = A; break; }
      A += c;
    }
    su[1] = (uint32_t)sel;
    sf[1] = T - Asel;
  }
  __syncthreads();
}

__global__ __launch_bounds__(TPB) void TopKTopPSampler_81870666596473_kernel(
    const float* __restrict__ logits, const float* __restrict__ temps,
    const int* __restrict__ karr, const float* __restrict__ parr,
    float* __restrict__ out) {
  const int row = blockIdx.x;
  const int t = threadIdx.x;
  const float invT = 1.0f / temps[row];
  const uint32_t kk = (uint32_t)karr[row];
  const float p = parr[row];
  const float* rowp = logits + (size_t)row * VV;
  float* orow = out + (size_t)row * VV;

  __shared__ float sbuf[2 * CHUNK];    // 16 KB async staging, double buffered
  __shared__ uint32_t hist[NB];        // 8 KB, reused as float bins
  __shared__ uint32_t gsm[64];
  __shared__ uint32_t su[4];
  __shared__ float sf[4];
  float* fhist = reinterpret_cast<float*>(hist);
  float* gsmf = reinterpret_cast<float*>(gsm);

  // ---------- Pass 1: row max + count histogram over key[31:21] ----------
  for (int i = t; i < NB; i += TPB) hist[i] = 0;
  if (t == 0) su[0] = 0;
  __syncthreads();
  uint32_t lmax = 0;
  stream_row(rowp, invT, sbuf, [&](float x, int) {
    uint32_t key = f2key(x);
    lmax = lmax > key ? lmax : key;
    atomicAdd(&hist[key >> 21], 1u);
  });
  atomicMax(&su[0], lmax);
  __syncthreads();
  cross_count(hist, NB, kk, gsm, su);
  const uint32_t sel1 = su[1];
  const uint32_t rem1 = su[2];
  const float mx = key2f(su[0]);

  // ---------- Pass 2: count histogram over key[20:10] within sel1 ----------
  for (int i = t; i < NB; i += TPB) hist[i] = 0;
  __syncthreads();
  stream_row(rowp, invT, sbuf, [&](float x, int) {
    uint32_t key = f2key(x);
    if ((key >> 21) == sel1) atomicAdd(&hist[(key >> 10) & 0x7FFu], 1u);
  });
  __syncthreads();
  cross_count(hist, NB, rem1, gsm, su);
  const uint32_t sel2 = su[1];
  const uint32_t rem2 = su[2];

  // ---------- Pass 3: count histogram over key[9:0] -> exact top-k key ----
  for (int i = t; i < NB3; i += TPB) hist[i] = 0;
  __syncthreads();
  const uint32_t pre12 = (sel1 << 11) | sel2;
  stream_row(rowp, invT, sbuf, [&](float x, int) {
    uint32_t key = f2key(x);
    if ((key >> 10) == pre12) atomicAdd(&hist[key & 0x3FFu], 1u);
  });
  __syncthreads();
  cross_count(hist, NB3, rem2, gsm, su);
  const uint32_t thrKey = (sel1 << 21) | (sel2 << 10) | su[1];

  // ---------- Pass 4: weighted (exp) histogram over key[31:21] ------------
  // keep x iff (sum of exp over strictly-larger kept elems) < p * S
  for (int i = t; i < NB; i += TPB) hist[i] = 0;
  __syncthreads();
  stream_row(rowp, invT, sbuf, [&](float x, int) {
    uint32_t key = f2key(x);
    if (key >= thrKey) atomicAdd(&fhist[key >> 21], __expf(x - mx));
  });
  __syncthreads();
  cross_weight(fhist, NB, true, p, 0.f, gsmf, su, sf);
  const uint32_t w1 = su[1];
  const float wrem1 = sf[1];

  // ---------- Pass 5: weighted histogram over key[20:10] within w1 --------
  for (int i = t; i < NB; i += TPB) hist[i] = 0;
  __syncthreads();
  stream_row(rowp, invT, sbuf, [&](float x, int) {
    uint32_t key = f2key(x);
    if (key >= thrKey && (key >> 21) == w1)
      atomicAdd(&fhist[(key >> 10) & 0x7FFu], __expf(x - mx));
  });
  __syncthreads();
  cross_weight(fhist, NB, false, 0.f, wrem1, gsmf, su, sf);
  const uint32_t w2 = su[1];
  const float wrem2 = sf[1];

  // ---------- Pass 6: weighted histogram over key[9:0] -> top-p key -------
  for (int i = t; i < NB3; i += TPB) hist[i] = 0;
  __syncthreads();
  const uint32_t wpre12 = (w1 << 11) | w2;
  stream_row(rowp, invT, sbuf, [&](float x, int) {
    uint32_t key = f2key(x);
    if (key >= thrKey && (key >> 10) == wpre12)
      atomicAdd(&fhist[key & 0x3FFu], __expf(x - mx));
  });
  __syncthreads();
  cross_weight(fhist, NB3, false, 0.f, wrem2, gsmf, su, sf);
  const uint32_t vKey = (w1 << 21) | (w2 << 10) | su[1];  // >= thrKey by constr.

  // ---------- Pass 7: write masked output ---------------------------------
  const float ninf = __uint_as_float(0xFF800000u);
  stream_row(rowp, invT, sbuf, [&](float x, int idx) {
    uint32_t key = f2key(x);
    orow[idx] = (key >= vKey) ? x : ninf;
  });
}

extern "C" void kernel_launch(void* const* d_in, const int* in_sizes, int n_in,
                              void* d_out, int out_size, void* d_ws,
                              size_t ws_size, hipStream_t stream) {
  (void)n_in; (void)out_size; (void)d_ws; (void)ws_size;
  const float* logits = (const float*)d_in[0];
  const float* temps = (const float*)d_in[1];
  const int* karr = (const int*)d_in[2];
  const float* parr = (const float*)d_in[3];
  float* out = (float*)d_out;
  const int B = in_sizes[1];  // 256 rows (one workgroup per row)
  hipLaunchKernelGGL(TopKTopPSampler_81870666596473_kernel, dim3(B), dim3(TPB),
                     0, stream, logits, temps, karr, parr, out);
}